// Wealth_21182778704643
// MI455X (gfx1250) — compile-verified
//
#include <hip/hip_runtime.h>
#include <stdint.h>

typedef float v2f __attribute__((ext_vector_type(2)));
typedef float v8f __attribute__((ext_vector_type(8)));

#define DT_ (1.0f / 252.0f)
#define GAMMA_ 0.1f
#define B_ 64
#define T_ 512
#define N_ 64
#define M_ 64

// One 128-thread workgroup per (b,t).
//  Phase 1: all 4 waves async-copy the 16KB sigma[b,t,:,:] tile global->LDS
//           (global_load_async_to_lds_b128, ASYNCcnt), stage action row.
//  Phase 2: wave w computes risk[m0..m0+15] (m0 = 16w) with
//           V_WMMA_F32_16X16X4_F32 accumulating over 16 n-chunks.
//           A rows are all identical (= action chunk), so every D row equals
//           the risk row and we read element 0 of the accumulator.
//  Phase 3: (risk+zeta)^2 wave-reduction; wave 0 computes action.mu wealth dot.
//  partial[b*T+t] = DT*rw - (GAMMA/2)*DT*sum_m (risk+zeta)^2
__global__ __launch_bounds__(128) void wealth_partial_kernel(
    const float* __restrict__ action,   // (B,T,N)
    const float* __restrict__ mu,       // (B,N,T)
    const float* __restrict__ sigma,    // (B,T,N,M)
    const float* __restrict__ zeta,     // (B,T,M)
    const float* __restrict__ bt,       // (T,1)
    float* __restrict__ partial)        // (B*T)
{
    __shared__ float sSig[N_ * M_];     // 16 KB sigma tile, row-major [n][m]
    __shared__ float sAct[N_];
    __shared__ float sRed[4];
    __shared__ float sRw;

    const int bt_idx = blockIdx.x;      // b*512 + t
    const int b   = bt_idx >> 9;
    const int t   = bt_idx & 511;
    const int tid = threadIdx.x;
    const int lane = tid & 31;
    const int wave = tid >> 5;

    // ---------------- Phase 1: async sigma tile -> LDS ----------------
    {
        const char* gbase = (const char*)(sigma + (size_t)bt_idx * (N_ * M_));
        unsigned ldsbase = (unsigned)(uintptr_t)(&sSig[0]);
#pragma unroll
        for (int j = 0; j < 8; ++j) {
            unsigned off = (unsigned)(tid + j * 128) * 16u;   // 16B per lane per op
            uint64_t ga = (uint64_t)(uintptr_t)(gbase + off);
            unsigned la = ldsbase + off;
            asm volatile("global_load_async_to_lds_b128 %0, %1, off"
                         :: "v"(la), "v"(ga) : "memory");
        }
    }
    if (tid < N_) sAct[tid] = action[(size_t)bt_idx * N_ + tid];

    asm volatile("s_wait_asynccnt 0x0" ::: "memory");
    __syncthreads();

    // ---------------- Phase 2: WMMA over n-chunks ----------------------
    const int m0 = wave * 16;
    const int h  = lane >> 4;            // half-wave selects K pair {0,1} vs {2,3}
    const int mc = m0 + (lane & 15);

    v8f c = {0.f, 0.f, 0.f, 0.f, 0.f, 0.f, 0.f, 0.f};
#pragma unroll
    for (int ck = 0; ck < 16; ++ck) {
        const int n0 = 4 * ck + 2 * h;
        v2f a;                            // A[*, k]: rows identical = action chunk
        a.x = sAct[n0];
        a.y = sAct[n0 + 1];
        v2f bb;                           // B[k, m] tile from LDS
        bb.x = sSig[n0 * M_ + mc];
        bb.y = sSig[(n0 + 1) * M_ + mc];
        c = __builtin_amdgcn_wmma_f32_16x16x4_f32(false, a, false, bb,
                                                  (short)0, c, false, false);
    }

    // ---------------- Phase 3: epilogue reductions ----------------------
    float risk = c[0];                                   // rows identical
    float r = risk + zeta[(size_t)bt_idx * M_ + mc];
    float s = r * r;                                     // each m counted twice
#pragma unroll
    for (int off2 = 16; off2 > 0; off2 >>= 1)
        s += __shfl_down(s, off2, 32);
    if (lane == 0) sRed[wave] = s;

    if (wave == 0) {                                     // wealth dot: action . mu[:,t]
        const size_t mu_base = ((size_t)b * N_) * T_ + (size_t)t;
        float rw = sAct[lane]      * mu[mu_base + (size_t)lane * T_]
                 + sAct[lane + 32] * mu[mu_base + (size_t)(lane + 32) * T_];
#pragma unroll
        for (int off2 = 16; off2 > 0; off2 >>= 1)
            rw += __shfl_down(rw, off2, 32);
        if (lane == 0) sRw = rw + bt[t];
    }
    __syncthreads();

    if (tid == 0) {
        float sq = 0.5f * (sRed[0] + sRed[1] + sRed[2] + sRed[3]);
        partial[bt_idx] = DT_ * sRw - (0.5f * GAMMA_) * DT_ * sq;
    }
}

// Deterministic fixed-order reduction over T per batch element.
__global__ __launch_bounds__(128) void wealth_reduce_kernel(
    const float* __restrict__ partial, float* __restrict__ out)
{
    __shared__ float red[128];
    const int b = blockIdx.x;
    const int tid = threadIdx.x;
    float s = 0.f;
#pragma unroll
    for (int i = 0; i < 4; ++i)
        s += partial[b * T_ + tid + i * 128];
    red[tid] = s;
    __syncthreads();
    for (int off = 64; off > 0; off >>= 1) {
        if (tid < off) red[tid] += red[tid + off];
        __syncthreads();
    }
    if (tid == 0) out[b] = red[0];
}

extern "C" void kernel_launch(void* const* d_in, const int* in_sizes, int n_in,
                              void* d_out, int out_size, void* d_ws, size_t ws_size,
                              hipStream_t stream)
{
    const float* action = (const float*)d_in[0];
    const float* mu     = (const float*)d_in[1];
    const float* sigma  = (const float*)d_in[2];
    const float* zeta   = (const float*)d_in[3];
    const float* bt     = (const float*)d_in[4];

    float* partial = (float*)d_ws;   // B*T floats = 128 KB

    wealth_partial_kernel<<<B_ * T_, 128, 0, stream>>>(action, mu, sigma, zeta,
                                                       bt, partial);
    wealth_reduce_kernel<<<B_, 128, 0, stream>>>(partial, (float*)d_out);
}